// SCOUT_MTP_69423851372803
// MI455X (gfx1250) — compile-verified
//
#include <hip/hip_runtime.h>
#include <hip/hip_bf16.h>
#include <math.h>

typedef __attribute__((ext_vector_type(2))) float v2f;
typedef __attribute__((ext_vector_type(8))) float v8f;

#define N_NODES   50000
#define N_EDGES   800000
#define IN_FEATS  128
#define OUT_FEATS 128
#define HEADS     2
#define NCOL      256            // HEADS*OUT_FEATS
#define ROW_TILES 3125           // 50000 / 16
#define NEG_SLOPE 0.2f

// LDS layout for GEMM kernel (dynamic LDS):
//   Bswz: fragment-major B. chunk c = (k>>1) + hf  (c in 0..63);
//         Bswz[c*B_PITCH2 + n] : float2 = { W[n*128 + 2c], W[n*128 + 2c + 1] }
//         -> per-lane B fragment is ONE ds_load_b64 into an even VGPR pair.
//   A0/A1: double-buffered A strips [16][A_STRIDE], filled by
//          global_load_async_to_lds_b128 (ASYNCcnt-tracked DMA path).
//          A_STRIDE = 132 -> row pitch 528B = 33*16 keeps b128 targets aligned.
#define B_PITCH2  273
#define A_STRIDE  132
#define A_BUF_FLOATS (16 * A_STRIDE)
#define LDS_BYTES ((64 * B_PITCH2 * 2 + 2 * A_BUF_FLOATS) * 4)

#define ORD_NEG_INF 0x007FFFFFu  // f2ord(-inf)

__device__ __forceinline__ unsigned f2ord(float f) {
  unsigned u = __float_as_uint(f);
  return (u & 0x80000000u) ? ~u : (u | 0x80000000u);
}
__device__ __forceinline__ float ord2f(unsigned o) {
  unsigned u = (o & 0x80000000u) ? (o & 0x7FFFFFFFu) : ~o;
  return __uint_as_float(u);
}

// ---------------- K0: init workspace & output -------------------------------
__global__ void k0_init(float* __restrict__ out, float* __restrict__ denom,
                        unsigned* __restrict__ emax_ord) {
  const long long total = (long long)N_NODES * NCOL;
  for (long long i = (long long)blockIdx.x * blockDim.x + threadIdx.x;
       i < total; i += (long long)gridDim.x * blockDim.x) {
    out[i] = 0.0f;
    if (i < (long long)N_NODES * HEADS) {
      denom[i] = 0.0f;
      emax_ord[i] = ORD_NEG_INF;
    }
  }
}

// Issue async memory->LDS copy of one 16x128 A strip (2 b128 ops per thread).
__device__ __forceinline__ void issue_async_A(const float* __restrict__ feat,
                                              int row0, float* Abuf,
                                              int tid, int nthreads) {
  for (int idx = tid; idx < 16 * 32; idx += nthreads) {
    const int m  = idx >> 5;
    const int kc = (idx & 31) * 4;                       // float4 index
    const float* g = feat + (size_t)(row0 + m) * IN_FEATS + kc;
    const unsigned loff = (unsigned)(uintptr_t)(&Abuf[m * A_STRIDE + kc]);
    asm volatile("global_load_async_to_lds_b128 %0, %1, off"
                 :: "v"(loff), "v"(g) : "memory");
  }
}

// ---------------- K1: ft = feat x W^T via V_WMMA_F32_16X16X4_F32 ------------
// Persistent blocks; B staged once per block in fragment-major layout.
// A strips double-buffered through LDS with global_load_async_to_lds_b128,
// next strip prefetched while current strip runs on the matrix pipe.
// 8 waves/block; each wave owns TWO 16x16 column tiles (shared A fragment,
// two independent accumulator chains), fragment loads pipelined one k-step.
__global__ __launch_bounds__(256) void k1_gemm_wmma(
    const float* __restrict__ feat, const float* __restrict__ W,
    float* __restrict__ ft) {
  extern __shared__ float lds[];
  v2f*   Bswz = (v2f*)lds;                    // 64 chunks x B_PITCH2 float2
  float* A0   = lds + 64 * B_PITCH2 * 2;      // [16][A_STRIDE]
  float* A1   = A0 + A_BUF_FLOATS;            // [16][A_STRIDE]

  const int tid = threadIdx.x;
  const int nthreads = blockDim.x;

  // ---- Stage B fragment-major: thread idx -> (n = idx>>6, c = idx&63) ----
  // k0 = 2*c, so 64 consecutive threads sweep one W row contiguously.
  for (int idx = tid; idx < NCOL * 64; idx += nthreads) {
    const int n = idx >> 6;
    const int c = idx & 63;
    const float2 w2 = *(const float2*)(W + (size_t)n * IN_FEATS + 2 * c);
    v2f b; b.x = w2.x; b.y = w2.y;
    Bswz[c * B_PITCH2 + n] = b;
  }

  const int lane = tid & 31;
  const int wave = tid >> 5;             // 0..7 -> pair of column tiles
  const int hf   = lane >> 4;            // half-wave select
  const int l    = lane & 15;
  const int col0 = wave * 32;            // tile0 cols [col0, col0+16)
  const int col1 = col0 + 16;            // tile1 cols [col1, col1+16)

  // Prefetch first strip into A0.
  int strip = blockIdx.x;
  if (strip < ROW_TILES) issue_async_A(feat, strip * 16, A0, tid, nthreads);

  for (int it = 0; strip < ROW_TILES; strip += gridDim.x, ++it) {
    float* Abuf  = (it & 1) ? A1 : A0;
    float* Anext = (it & 1) ? A0 : A1;
    const int nstrip = strip + gridDim.x;
    if (nstrip < ROW_TILES) {
      issue_async_A(feat, nstrip * 16, Anext, tid, nthreads);
      // drain only the previous strip's 2 async ops; keep prefetch in flight
      asm volatile("s_wait_asynccnt 0x2" ::: "memory");
    } else {
      asm volatile("s_wait_asynccnt 0x0" ::: "memory");
    }
    __syncthreads();                     // Abuf complete for all waves

    const int row0 = strip * 16;
    v8f c0 = {0.f, 0.f, 0.f, 0.f, 0.f, 0.f, 0.f, 0.f};
    v8f c1 = {0.f, 0.f, 0.f, 0.f, 0.f, 0.f, 0.f, 0.f};

    // prologue: fragments for k = 0 (chunk c = hf)
    v2f a, b0, b1;
    {
      const float* ap = Abuf + l * A_STRIDE + 2 * hf;
      a.x = ap[0]; a.y = ap[1];
      b0 = Bswz[hf * B_PITCH2 + col0 + l];
      b1 = Bswz[hf * B_PITCH2 + col1 + l];
    }

    #pragma unroll
    for (int k = 0; k < IN_FEATS; k += 4) {
      // prefetch k+4 fragments (wraps to 0 on last iter; harmless valid reads)
      const int kn = (k + 4) & (IN_FEATS - 1);
      const int cc = (kn >> 1) + hf;     // fragment chunk for next step
      const float* ap = Abuf + l * A_STRIDE + kn + 2 * hf;
      v2f an, b0n, b1n;
      an.x = ap[0]; an.y = ap[1];
      b0n = Bswz[cc * B_PITCH2 + col0 + l];
      b1n = Bswz[cc * B_PITCH2 + col1 + l];

      c0 = __builtin_amdgcn_wmma_f32_16x16x4_f32(false, a, false, b0,
                                                 (short)0, c0, false, false);
      c1 = __builtin_amdgcn_wmma_f32_16x16x4_f32(false, a, false, b1,
                                                 (short)0, c1, false, false);
      a = an; b0 = b0n; b1 = b1n;
    }

    // ---- Store C tiles: VGPR r holds (M = r + 8*hf, N = l) ----
    #pragma unroll
    for (int r = 0; r < 8; ++r) {
      const size_t row = (size_t)(row0 + r + 8 * hf);
      ft[row * NCOL + col0 + l] = c0[r];
      ft[row * NCOL + col1 + l] = c1[r];
    }
    __syncthreads();   // all reads of Abuf done before next iter overwrites it
  }
}

// ---------------- K2: el/er per node (wave per node) ------------------------
__global__ void k2_attn_dots(const float* __restrict__ ft,
                             const float* __restrict__ attn_l,
                             const float* __restrict__ attn_r,
                             float* __restrict__ el, float* __restrict__ er) {
  const int wid  = (blockIdx.x * blockDim.x + threadIdx.x) >> 5;
  const int lane = threadIdx.x & 31;
  if (wid >= N_NODES) return;
  const int h = lane >> 4;               // lanes 0-15 -> head 0, 16-31 -> head 1
  const int obase = (lane * 8) & 127;
  const float4* p = (const float4*)(ft + (size_t)wid * NCOL + lane * 8);
  const float4 v0 = p[0], v1 = p[1];
  const float* al = attn_l + h * OUT_FEATS + obase;
  const float* ar = attn_r + h * OUT_FEATS + obase;
  float sl = v0.x * al[0] + v0.y * al[1] + v0.z * al[2] + v0.w * al[3] +
             v1.x * al[4] + v1.y * al[5] + v1.z * al[6] + v1.w * al[7];
  float sr = v0.x * ar[0] + v0.y * ar[1] + v0.z * ar[2] + v0.w * ar[3] +
             v1.x * ar[4] + v1.y * ar[5] + v1.z * ar[6] + v1.w * ar[7];
  #pragma unroll
  for (int m = 1; m < 16; m <<= 1) {     // reduce within each 16-lane half
    sl += __shfl_xor(sl, m, 32);
    sr += __shfl_xor(sr, m, 32);
  }
  if ((lane & 15) == 0) {
    el[wid * HEADS + h] = sl;
    er[wid * HEADS + h] = sr;
  }
}

// ---------------- K3: edge score + leaky_relu + segment max -----------------
__global__ void k3_edge_score(const int* __restrict__ src, const int* __restrict__ dst,
                              const float* __restrict__ e_w,
                              const float* __restrict__ attn_ew,
                              const float* __restrict__ el, const float* __restrict__ er,
                              float* __restrict__ escore, unsigned* __restrict__ emax_ord) {
  const int idx = blockIdx.x * blockDim.x + threadIdx.x;
  if (idx >= N_EDGES * HEADS) return;
  const int e = idx >> 1, h = idx & 1;
  const int s = src[e], d = dst[e];
  float sc = el[s * HEADS + h] + er[d * HEADS + h] +
             e_w[e * 2 + 0] * attn_ew[h * 2 + 0] +
             e_w[e * 2 + 1] * attn_ew[h * 2 + 1];
  sc = sc > 0.0f ? sc : NEG_SLOPE * sc;
  escore[idx] = sc;
  atomicMax(&emax_ord[d * HEADS + h], f2ord(sc));
}

// ---------------- K4: exp(e - max) + segment sum ----------------------------
__global__ void k4_edge_exp(const int* __restrict__ dst,
                            const unsigned* __restrict__ emax_ord,
                            float* __restrict__ escore, float* __restrict__ denom) {
  const int idx = blockIdx.x * blockDim.x + threadIdx.x;
  if (idx >= N_EDGES * HEADS) return;
  const int e = idx >> 1, h = idx & 1;
  const int d = dst[e];
  float m = ord2f(emax_ord[d * HEADS + h]);
  if (!isfinite(m)) m = 0.0f;
  const float ee = expf(escore[idx] - m);
  escore[idx] = ee;                       // now holds exp value
  atomicAdd(&denom[d * HEADS + h], ee);
}

// ---------------- K5: aggregate  out[dst] += ft[src] * a  (wave/edge-head) --
__global__ void k5_aggregate(const int* __restrict__ src, const int* __restrict__ dst,
                             const float* __restrict__ escore,
                             const float* __restrict__ denom,
                             const float* __restrict__ ft, float* __restrict__ out) {
  const long long wid = ((long long)blockIdx.x * blockDim.x + threadIdx.x) >> 5;
  const int lane = threadIdx.x & 31;
  if (wid >= (long long)N_EDGES * HEADS) return;
  const int e = (int)(wid >> 1), h = (int)(wid & 1);
  const int s = src[e], d = dst[e];
  float dn = denom[d * HEADS + h];
  dn = dn > 0.0f ? dn : 1.0f;
  const float a = escore[e * HEADS + h] / dn;
  const int base = h * OUT_FEATS + lane * 4;
  const float4 v = *(const float4*)(ft + (size_t)s * NCOL + base);
  float* o = out + (size_t)d * NCOL + base;
  atomicAdd(o + 0, v.x * a);
  atomicAdd(o + 1, v.y * a);
  atomicAdd(o + 2, v.z * a);
  atomicAdd(o + 3, v.w * a);
}

// ---------------- K6: residual + ELU ----------------------------------------
__global__ void k6_finalize(const float* __restrict__ feat, float* __restrict__ out) {
  const long long idx = (long long)blockIdx.x * blockDim.x + threadIdx.x;
  if (idx >= (long long)N_NODES * NCOL) return;
  const int n = (int)(idx >> 8);
  const int o = (int)(idx & 127);        // feature index within head
  const float x = out[idx] + feat[(size_t)n * IN_FEATS + o];
  out[idx] = x > 0.0f ? x : expm1f(x);
}

// ---------------- launch -----------------------------------------------------
extern "C" void kernel_launch(void* const* d_in, const int* in_sizes, int n_in,
                              void* d_out, int out_size, void* d_ws, size_t ws_size,
                              hipStream_t stream) {
  const float* feat    = (const float*)d_in[0];
  const float* e_w     = (const float*)d_in[1];
  const int*   src     = (const int*)d_in[2];
  const int*   dst     = (const int*)d_in[3];
  const float* W       = (const float*)d_in[4];
  const float* attn_l  = (const float*)d_in[5];
  const float* attn_r  = (const float*)d_in[6];
  const float* attn_ew = (const float*)d_in[7];
  float* out = (float*)d_out;

  char* ws = (char*)d_ws;
  float*    ft       = (float*)(ws);                       // 50000*256 f32
  float*    escore   = (float*)(ws + 51200000);            // 800000*2 f32
  float*    el       = (float*)(ws + 57600000);            // 50000*2 f32
  float*    er       = (float*)(ws + 58000000);            // 50000*2 f32
  float*    denom    = (float*)(ws + 58400000);            // 50000*2 f32
  unsigned* emax_ord = (unsigned*)(ws + 58800000);         // 50000*2 u32

  k0_init<<<2048, 256, 0, stream>>>(out, denom, emax_ord);
  k1_gemm_wmma<<<512, 256, LDS_BYTES, stream>>>(feat, W, ft);
  k2_attn_dots<<<(N_NODES * 32 + 255) / 256, 256, 0, stream>>>(ft, attn_l, attn_r, el, er);
  k3_edge_score<<<(N_EDGES * HEADS + 255) / 256, 256, 0, stream>>>(
      src, dst, e_w, attn_ew, el, er, escore, emax_ord);
  k4_edge_exp<<<(N_EDGES * HEADS + 255) / 256, 256, 0, stream>>>(dst, emax_ord, escore, denom);
  {
    const long long waves = (long long)N_EDGES * HEADS;
    const long long blocks = (waves * 32 + 255) / 256;
    k5_aggregate<<<(int)blocks, 256, 0, stream>>>(src, dst, escore, denom, ft, out);
  }
  k6_finalize<<<(int)(((long long)N_NODES * NCOL + 255) / 256), 256, 0, stream>>>(feat, out);
}